// ExcitationGeneration_47502338294557
// MI455X (gfx1250) — compile-verified
//
#include <hip/hip_runtime.h>

#define BB 16
#define NN 8000
#define PP 80
#define LL (NN * PP)          // 640000 samples per row
#define TILE 1024
#define NTILES (LL / TILE)    // 625 tiles per row

typedef unsigned int u32x4 __attribute__((ext_vector_type(4)));
typedef int          i32x8 __attribute__((ext_vector_type(8)));
typedef int          i32x4 __attribute__((ext_vector_type(4)));

// ---- per-sample pitch interpolation / q / reset-flag ------------------------
// mask[n] = p[n] > 0 (base_mask from ORIGINAL p); adjusted p only differs at
// the first unvoiced frame after a voiced run (p_adj[n] = p[n-1]) which only
// matters as the *right* interpolation endpoint of the last voiced frame.
__device__ __forceinline__ void elem_qf(const float* __restrict__ prow, int t,
                                        float& pi, float& qf, int& f) {
  int n = t / PP;
  int k = t - n * PP;
  float pn = prow[n];
  pi = 0.0f;
  if (pn > 0.0f) {                       // voiced frame
    float b;
    if (n + 1 < NN) {
      float p1 = prow[n + 1];
      // padj(n+1): voiced->unvoiced transition carries previous pitch
      b = (p1 == 0.0f && pn >= 1.0f) ? pn : p1;
    } else {
      b = pn;                            // replicate-pad last frame
    }
    float frac = (float)k / (float)PP;
    pi = pn * (1.0f - frac) + b * frac;
  }
  qf = (pi > 0.0f) ? (1.0f / pi) : 0.0f;
  f  = (pn > 0.0f) ? 0 : 1;              // reset flag at unvoiced samples
}

// ---- wave32 segmented inclusive scan (Hillis-Steele over lanes) -------------
__device__ __forceinline__ void wave_seg_scan(double& q, int& f, int lane) {
#pragma unroll
  for (int d = 1; d < 32; d <<= 1) {
    double qp = __shfl_up(q, d, 32);
    int    fp = __shfl_up(f, d, 32);
    if (lane >= d) {
      if (!f) q += qp;
      f |= fp;
    }
  }
}

// ---- pass 1: per-tile segmented aggregates (fully parallel) -----------------
__global__ __launch_bounds__(TILE) void k_tile_reduce(
    const float* __restrict__ p, const float* __restrict__ noise,
    double* __restrict__ aggQ, unsigned* __restrict__ aggF) {
  __shared__ double s_wq[32];
  __shared__ int    s_wf[32];
  int tid = threadIdx.x, lane = tid & 31, wid = tid >> 5;
  int bx = blockIdx.x, row = blockIdx.y;
  int t = bx * TILE + tid;

  // Warm the 192MB L2 with the noise row: pass 3 then reads noise from L2.
  __builtin_prefetch(noise + (size_t)row * LL + t, 0, 1);

  const float* prow = p + row * NN;
  float pi, qf; int f;
  elem_qf(prow, t, pi, qf, f);
  double q = (double)qf;

  wave_seg_scan(q, f, lane);
  if (lane == 31) { s_wq[wid] = q; s_wf[wid] = f; }
  __syncthreads();
  if (wid == 0) {
    double wq = s_wq[lane]; int wf = s_wf[lane];
    wave_seg_scan(wq, wf, lane);
    if (lane == 31) {                     // tile aggregate (in-order combine)
      aggQ[row * NTILES + bx] = wq;
      aggF[row * NTILES + bx] = (unsigned)wf;
    }
  }
}

// ---- pass 2: scan 625 tile aggregates per row (one wave per row) ------------
// pref[j] = phase value at the last sample BEFORE tile j (0 for j==0).
__global__ __launch_bounds__(32) void k_scan_partials(
    const double* __restrict__ aggQ, const unsigned* __restrict__ aggF,
    double* __restrict__ pref) {
  int lane = threadIdx.x;
  int row = blockIdx.x;
  double carry = 0.0;   // carry flag is irrelevant: combine never reads lhs flag
  const int chunks = (NTILES + 31) / 32;
  for (int c = 0; c < chunks; ++c) {
    int j = c * 32 + lane;
    double q = 0.0; int f = 0;                          // identity padding
    if (j < NTILES) { q = aggQ[row * NTILES + j]; f = (int)aggF[row * NTILES + j]; }
    wave_seg_scan(q, f, lane);
    double exQ = __shfl_up(q, 1, 32);
    int    exF = __shfl_up(f, 1, 32);
    if (lane == 0) { exQ = 0.0; exF = 0; }
    double Pj = exF ? exQ : carry + exQ;
    if (j < NTILES) pref[row * NTILES + j] = Pj;
    double aQ = __shfl(q, 31, 32);
    int    aF = __shfl(f, 31, 32);
    carry = aF ? aQ : carry + aQ;                       // chunk carry
  }
}

// ---- pass 3: recompute in-tile scan, detect pulses, merge noise -------------
// Noise tile (4KB) is staged into LDS by BOTH CDNA5 async engines:
//   * first 512 floats: one Tensor Data Mover descriptor issued by wave 0
//     (TENSORcnt; waited by wave 0 before the block barrier)
//   * last  512 floats: per-lane global_load_async_to_lds_b32 (ASYNCcnt;
//     each wave reads only slots it loaded itself, so a wave-local wait works)
__global__ __launch_bounds__(TILE) void k_apply(
    const float* __restrict__ p, const float* __restrict__ noise,
    const double* __restrict__ pref, float* __restrict__ out) {
  __shared__ float  shnoise[TILE];
  __shared__ double s_wq[32]; __shared__ int s_wf[32];
  __shared__ double s_iq[32]; __shared__ int s_if[32];
  int tid = threadIdx.x, lane = tid & 31, wid = tid >> 5;
  int bx = blockIdx.x, row = blockIdx.y;
  int t = bx * TILE + tid;

  const float* nrow = noise + (size_t)row * (size_t)LL;

  if (wid == 0) {
    // --- TDM: DMA noise[tile .. tile+511] -> shnoise[0..511] ---------------
    unsigned long long gaddr = (unsigned long long)(nrow + (size_t)bx * TILE);
    unsigned lds_base = (unsigned)(unsigned long long)(&shnoise[0]);
    u32x4 g0;
    g0.x = 1u;                                            // count=1, user D#
    g0.y = lds_base;                                      // lds_addr [63:32]
    g0.z = (unsigned)(gaddr & 0xFFFFFFFFu);               // global_addr[31:0]
    g0.w = (unsigned)((gaddr >> 32) & 0x01FFFFFFu)        // global_addr[56:32]
           | (2u << 30);                                  // type = 2 (image)
    i32x8 g1;
    g1[0] = 0x00020000;                 // workgroup_mask=0, data_size=2 (4B)
    g1[1] = (int)(512u << 16);          // tensor_dim0[15:0]=512 in bits[31:16]
    g1[2] = (int)(1u << 16);            // tensor_dim0 hi=0; tensor_dim1=1
    g1[3] = (int)(512u << 16);          // tensor_dim1 hi=0; tile_dim0=512
    g1[4] = 1;                          // tile_dim1=1, tile_dim2=0
    g1[5] = 512;                        // tensor_dim0_stride = 512
    g1[6] = 0;                          // stride hi / dim1_stride lo
    g1[7] = 0;
    i32x4 z4 = {0, 0, 0, 0};            // groups 2/3: tensor is 2D
    i32x8 z8 = {0, 0, 0, 0, 0, 0, 0, 0};// extra group (6-arg builtin), unused
    __builtin_amdgcn_tensor_load_to_lds(g0, g1, z4, z4, z8, 0);
  } else if (tid >= TILE / 2) {
    // --- per-lane async global->LDS for the second half of the tile --------
    unsigned lds_off = (unsigned)(unsigned long long)(&shnoise[tid]);
    unsigned voff = (unsigned)t * 4u;
    unsigned long long nbase = (unsigned long long)nrow;
    asm volatile("global_load_async_to_lds_b32 %0, %1, %2"
                 :: "v"(lds_off), "v"(voff), "s"(nbase) : "memory");
  }

  const float* prow = p + row * NN;
  float pi, qf; int f;
  elem_qf(prow, t, pi, qf, f);
  double q = (double)qf;

  wave_seg_scan(q, f, lane);                       // in-wave inclusive
  if (lane == 31) { s_wq[wid] = q; s_wf[wid] = f; }
  __syncthreads();
  if (wid == 0) {
    double wq = s_wq[lane]; int wf = s_wf[lane];
    wave_seg_scan(wq, wf, lane);                   // scan of wave aggregates
    s_iq[lane] = wq; s_if[lane] = wf;
    __builtin_amdgcn_s_wait_tensorcnt(0);          // TDM tile half is in LDS
  }
  __syncthreads();                                 // publishes s_iq + TDM data

  double exQ = 0.0; int exF = 0;                   // exclusive over prior waves
  if (wid > 0) { exQ = s_iq[wid - 1]; exF = s_if[wid - 1]; }
  double Pj = pref[row * NTILES + bx];             // phase before this tile
  double Ew = exF ? exQ : Pj + exQ;                // phase before this wave

  float phase = (float)(f ? q : Ew + q);           // reference casts to f32
  float pprev = __shfl_up(phase, 1, 32);           // previous sample's phase
  if (lane == 0) pprev = (float)Ew;

  float r  = __builtin_ceilf(phase);
  float rp = __builtin_ceilf(pprev);
  float e = ((r - rp) >= 1.0f) ? __builtin_sqrtf(pi) : 0.0f;

  __builtin_amdgcn_s_wait_asynccnt(0);             // own async slots landed
  asm volatile("" ::: "memory");                   // keep ds read after waits
  if (f) e = shnoise[tid];                         // unvoiced -> gauss noise

  out[(size_t)row * LL + t] = e;
}

extern "C" void kernel_launch(void* const* d_in, const int* in_sizes, int n_in,
                              void* d_out, int out_size, void* d_ws, size_t ws_size,
                              hipStream_t stream) {
  (void)in_sizes; (void)n_in; (void)out_size; (void)ws_size;
  const float* p     = (const float*)d_in[0];   // (16, 8000)
  const float* noise = (const float*)d_in[1];   // (16, 640000)
  float* out = (float*)d_out;                   // (16, 640000)

  char* ws = (char*)d_ws;                       // 200 KB used, laid out 8B-aligned
  double*   aggQ = (double*)ws;                                    // 80000 B
  unsigned* aggF = (unsigned*)(ws + (size_t)BB * NTILES * 8);      // 40000 B
  double*   pref = (double*)(ws + (size_t)BB * NTILES * 12);       // 80000 B

  dim3 grid(NTILES, BB);
  k_tile_reduce  <<<grid, TILE, 0, stream>>>(p, noise, aggQ, aggF);
  k_scan_partials<<<BB,   32,   0, stream>>>(aggQ, aggF, pref);
  k_apply        <<<grid, TILE, 0, stream>>>(p, noise, pref, out);
}